// MDH_75763223102178
// MI455X (gfx1250) — compile-verified
//
#include <hip/hip_runtime.h>

typedef __attribute__((ext_vector_type(2))) float v2f;
typedef __attribute__((ext_vector_type(8))) float v8f;

#define NHD   8      // heads
#define HDD   6      // head dim
#define DIMC  48
#define NT    27
#define GH    40
#define GW    40
#define GT    40
#define QSCALE 0.40824829046386301636f   // 6^-0.5

// B-matrix row t of OFFS(28x16): entries in {-1,0,1}, packed as 2 bits per column
// (code field value = entry + 1). Columns 3..15 and t>=27 are zero (field 01).
constexpr unsigned bcode(int t) {
    return (t >= NT)
        ? 0x55555555u                                     // all-zero row
        : (unsigned)(t / 9)                               // col0: i   (entry i-1)
        | ((unsigned)((t / 3) % 3) << 2)                  // col1: j
        | ((unsigned)(t % 3) << 4)                        // col2: l
        | 0x55555540u;                                    // cols 3..15 -> zero
}

__device__ __forceinline__ float bval(unsigned code, int col) {
    return (float)((int)((code >> (2 * col)) & 3u) - 1);
}

__global__ __launch_bounds__(256) void nat3d_wmma_kernel(
    const float* __restrict__ q,
    const float* __restrict__ k,
    const float* __restrict__ rpb,
    float* __restrict__ out)
{
    __shared__ float s_rpb[NHD * NT];
    int tid = threadIdx.x;
    if (tid < NHD * NT) s_rpb[tid] = rpb[tid];
    __syncthreads();

    const int lane = tid & 31;
    const int wave = tid >> 5;
    const int pair = blockIdx.x * 8 + wave;          // 0 .. 31999 (1600 columns * 20 z-pairs)
    const int z2   = pair % (GT / 2);
    const int tmp  = pair / (GT / 2);
    const int y    = tmp % GW;
    const int x    = tmp / GW;

    // 16 rows per wave: row m = 2 voxels (z2*2, z2*2+1) x 8 heads.
    // Lane m and lane m+16 both own row m (feeds WMMA f32 A layout directly).
    const int m    = lane & 15;
    const int half = lane >> 4;
    const int h    = m & 7;
    const int z    = z2 * 2 + (m >> 3);

    // ---- load q for this (voxel, head), pre-scaled ----
    const int lin = (x * GW + y) * GT + z;
    const float* qp = q + (size_t)lin * DIMC + h * HDD;
    float qv[HDD];
#pragma unroll
    for (int d = 0; d < HDD; d += 2) {
        v2f v = *(const v2f*)(qp + d);
        qv[d]     = v.x * QSCALE;
        qv[d + 1] = v.y * QSCALE;
    }

    // ---- 27 neighbor scores (zero-padded keys -> dot = 0, rpb still added) ----
    float p[NT];
    float mx = -1e30f;
#pragma unroll
    for (int t = 0; t < NT; ++t) {
        const int ti = t / 9, tj = (t / 3) % 3, tl = t % 3;
        const int nx = x + ti - 1, ny = y + tj - 1, nz = z + tl - 1;
        float dot = 0.0f;
        if ((unsigned)nx < GH && (unsigned)ny < GW && (unsigned)nz < GT) {
            const float* kp = k + (size_t)((nx * GW + ny) * GT + nz) * DIMC + h * HDD;
#pragma unroll
            for (int d = 0; d < HDD; d += 2) {
                v2f v = *(const v2f*)(kp + d);
                dot = fmaf(qv[d],     v.x, dot);
                dot = fmaf(qv[d + 1], v.y, dot);
            }
        }
        float s = dot + s_rpb[h * NT + t];
        p[t] = s;
        mx = fmaxf(mx, s);
    }

    // ---- softmax over the 27 scores ----
    float sum = 0.0f;
#pragma unroll
    for (int t = 0; t < NT; ++t) { p[t] = __expf(p[t] - mx); sum += p[t]; }
    const float inv = __builtin_amdgcn_rcpf(sum);   // fast v_rcp_f32, no IEEE div sequence
#pragma unroll
    for (int t = 0; t < NT; ++t) p[t] *= inv;

    // ---- out = P(16x28) x OFFS(28x16, cols 0..2 live) via 7x V_WMMA_F32_16X16X4_F32 ----
    // f32 A 16x4 layout: lanes 0-15 -> {v0:K=0, v1:K=1}, lanes 16-31 -> {v0:K=2, v1:K=3}.
    // Both half-lanes already hold row m's p[] -> no shuffles needed.
    const int col = lane & 15;
    v8f acc = {};
#pragma unroll
    for (int c = 0; c < 7; ++c) {
        const int t0 = 4 * c;
        const float a_lo0 = p[t0];
        const float a_lo1 = p[t0 + 1];
        const float a_hi0 = (t0 + 2 < NT) ? p[t0 + 2 < NT ? t0 + 2 : 0] : 0.0f;
        const float a_hi1 = (t0 + 3 < NT) ? p[t0 + 3 < NT ? t0 + 3 : 0] : 0.0f;
        v2f a, b;
        a.x = half ? a_hi0 : a_lo0;
        a.y = half ? a_hi1 : a_lo1;
        // f32 B 4x16 layout (mirror of A): lanes 0-15 -> rows K=t0,t0+1 ; lanes 16-31 -> K=t0+2,t0+3
        const unsigned cx = half ? bcode(t0 + 2) : bcode(t0);
        const unsigned cy = half ? bcode(t0 + 3) : bcode(t0 + 1);
        b.x = bval(cx, col);
        b.y = bval(cy, col);
        acc = __builtin_amdgcn_wmma_f32_16x16x4_f32(
            /*neg_a=*/false, a, /*neg_b=*/false, b,
            /*c_mod=*/(short)0, acc, /*reuse_a=*/false, /*reuse_b=*/false);
    }

    // ---- scatter D: lane holds column n=col, rows r + 8*half  (head=r, voxel z = 2*z2+half)
    // output layout: [24, 40, 40, 40] with channel c = h*3 + o
    if (col < 3) {
        const size_t sp = (size_t)((x * GW + y) * GT) + (size_t)(z2 * 2 + half);
#pragma unroll
        for (int r = 0; r < 8; ++r) {
            out[(size_t)(r * 3 + col) * (GH * GW * GT) + sp] = acc[r];
        }
    }
}

extern "C" void kernel_launch(void* const* d_in, const int* in_sizes, int n_in,
                              void* d_out, int out_size, void* d_ws, size_t ws_size,
                              hipStream_t stream) {
    (void)in_sizes; (void)n_in; (void)out_size; (void)d_ws; (void)ws_size;
    const float* q   = (const float*)d_in[0];
    const float* k   = (const float*)d_in[1];
    const float* rpb = (const float*)d_in[2];
    float* out = (float*)d_out;

    // 1600 (x,y) columns * 20 z-pairs = 32000 waves; 8 waves per 256-thread block
    dim3 grid(4000), block(256);
    nat3d_wmma_kernel<<<grid, block, 0, stream>>>(q, k, rpb, out);
}